// HypergraphLayer_58909771432740
// MI455X (gfx1250) — compile-verified
//
#include <hip/hip_runtime.h>

#define N_NODES   50000
#define N_EDGES   800000
#define DIM       64
#define NUM_GRAPHS 256

typedef __attribute__((ext_vector_type(2))) float v2f;
typedef __attribute__((ext_vector_type(8))) float v8f;

// ---- float atomic max via sign-aware int/uint atomics (monotone, init = -inf) ----
__device__ __forceinline__ void atomicMaxF32(float* addr, float val) {
    if (val >= 0.0f) {
        atomicMax((int*)addr, __float_as_int(val));
    } else {
        atomicMin((unsigned int*)addr, __float_as_uint(val));
    }
}

// ---- init: zero h1,h2 (contiguous 2*N*D floats) and set r0,r1,r2 (contiguous 3*G*D) to -inf
__global__ void hg_init_kernel(float* __restrict__ hzero, float* __restrict__ rinit) {
    unsigned i = blockIdx.x * blockDim.x + threadIdx.x;
    if (i < 2u * N_NODES * DIM) hzero[i] = 0.0f;
    if (i < 3u * NUM_GRAPHS * DIM) rinit[i] = __uint_as_float(0xFF800000u); // -inf
}

// ---- coef[e] = node_norm[src[e]] * node_norm[dst[e]] * edge_norm[e]
__global__ void hg_coef_kernel(const float* __restrict__ node_norm,
                               const float* __restrict__ edge_norm,
                               const int* __restrict__ src,
                               const int* __restrict__ dst,
                               float* __restrict__ coef) {
    int e = blockIdx.x * blockDim.x + threadIdx.x;
    if (e >= N_EDGES) return;
    coef[e] = node_norm[src[e]] * node_norm[dst[e]] * edge_norm[e];
}

// ---- per-graph max readout: 16 lanes per node, float4 per lane
__global__ void hg_readout_kernel(const float* __restrict__ h,
                                  const int* __restrict__ gids,
                                  float* __restrict__ r) {
    unsigned tid = blockIdx.x * blockDim.x + threadIdx.x;
    int n = (int)(tid >> 4);
    int part = (int)(tid & 15);
    if (n >= N_NODES) return;
    int g = gids[n];
    float4 v = *(const float4*)(h + (size_t)n * DIM + part * 4);
    float* base = r + (size_t)g * DIM + part * 4;
    atomicMaxF32(base + 0, v.x);
    atomicMaxF32(base + 1, v.y);
    atomicMaxF32(base + 2, v.z);
    atomicMaxF32(base + 3, v.w);
}

// ---- message scatter: h_next[dst] += h_prev[src] * coef, 16 lanes/edge, float4/lane
__global__ void hg_scatter_kernel(const float* __restrict__ hprev,
                                  const float* __restrict__ coef,
                                  const int* __restrict__ src,
                                  const int* __restrict__ dst,
                                  float* __restrict__ hnext) {
    unsigned tid = blockIdx.x * blockDim.x + threadIdx.x;
    int e = (int)(tid >> 4);
    int part = (int)(tid & 15);
    if (e >= N_EDGES) return;
    int s = src[e];
    int d = dst[e];
    float c = coef[e];
    float4 v = *(const float4*)(hprev + (size_t)s * DIM + part * 4);
    float* base = hnext + (size_t)d * DIM + part * 4;
    atomicAdd(base + 0, v.x * c);
    atomicAdd(base + 1, v.y * c);
    atomicAdd(base + 2, v.z * c);
    atomicAdd(base + 3, v.w * c);
}

// ---- epilogue: out = leaky_relu( mean(r0,r1,r2) @ W^T + b ) via V_WMMA_F32_16X16X4_F32
// One wave (32 threads) per 16x16 output tile; 16 row-tiles x 4 col-tiles = 64 blocks.
// A (16x4 f32):  lanes 0-15 hold rows M=0..15 with K={k0,k0+1}; lanes 16-31 K={k0+2,k0+3}
// B (4x16 f32):  VGPR v, lane-half h -> K = k0 + v + 2h, column N = lane&15
// C/D (16x16):   VGPR v -> M = v + 8*(lane>>4), N = lane&15
__global__ void hg_final_wmma_kernel(const float* __restrict__ r0,
                                     const float* __restrict__ r1,
                                     const float* __restrict__ r2,
                                     const float* __restrict__ W,
                                     const float* __restrict__ bias,
                                     float* __restrict__ out) {
    int lane = threadIdx.x & 31;
    int half = lane >> 4;
    int l16  = lane & 15;
    int tile_m = blockIdx.x >> 2;  // 0..15 (graph rows)
    int tile_n = blockIdx.x & 3;   // 0..3  (feature cols)

    int row  = tile_m * 16 + l16;  // A-matrix row carried by this lane
    int ncol = tile_n * 16 + l16;  // B-matrix column carried by this lane

    const float third = 1.0f / 3.0f;
    v8f acc = {};

    #pragma unroll
    for (int k0 = 0; k0 < DIM; k0 += 4) {
        int ka = k0 + 2 * half;
        v2f a, b;
        a.x = (r0[row * DIM + ka]     + r1[row * DIM + ka]     + r2[row * DIM + ka])     * third;
        a.y = (r0[row * DIM + ka + 1] + r1[row * DIM + ka + 1] + r2[row * DIM + ka + 1]) * third;
        // out = M @ W^T  =>  B[k][n] = W[n][k]
        b.x = W[ncol * DIM + ka];
        b.y = W[ncol * DIM + ka + 1];
        acc = __builtin_amdgcn_wmma_f32_16x16x4_f32(
            /*neg_a=*/false, a, /*neg_b=*/false, b,
            /*c_mod=*/(short)0, acc, /*reuse_a=*/false, /*reuse_b=*/false);
    }

    float bj = bias[ncol];
    #pragma unroll
    for (int v = 0; v < 8; ++v) {
        int g = tile_m * 16 + v + 8 * half;
        float val = acc[v] + bj;
        val = (val > 0.0f) ? val : 0.01f * val;
        out[(size_t)g * DIM + ncol] = val;
    }
}

extern "C" void kernel_launch(void* const* d_in, const int* in_sizes, int n_in,
                              void* d_out, int out_size, void* d_ws, size_t ws_size,
                              hipStream_t stream) {
    const float* x         = (const float*)d_in[0];   // [N, 64]
    const float* node_norm = (const float*)d_in[1];   // [N]
    const float* edge_norm = (const float*)d_in[2];   // [E]
    const float* W         = (const float*)d_in[3];   // [64, 64]
    const float* bvec      = (const float*)d_in[4];   // [64]
    const int*   src       = (const int*)d_in[5];     // [E]
    const int*   dst       = (const int*)d_in[6];     // [E]
    const int*   gids      = (const int*)d_in[7];     // [N] sorted
    float* out = (float*)d_out;                       // [256, 64]

    // workspace layout (all offsets 16B-aligned)
    char* ws = (char*)d_ws;
    float* coef = (float*)(ws + 0);                    //   800000 f
    float* h1   = (float*)(ws + 3200000);              //  3200000 f
    float* h2   = (float*)(ws + 16000000);             //  3200000 f  (contiguous after h1)
    float* r0   = (float*)(ws + 28800000);             //    16384 f
    float* r1   = (float*)(ws + 28800000 + 65536);     //    16384 f
    float* r2   = (float*)(ws + 28800000 + 131072);    //    16384 f

    // 1) init h1,h2 = 0 ; r0,r1,r2 = -inf   (6.4M threads covers both regions)
    hg_init_kernel<<<25000, 256, 0, stream>>>(h1, r0);

    // 2) edge coefficients
    hg_coef_kernel<<<(N_EDGES + 255) / 256, 256, 0, stream>>>(node_norm, edge_norm, src, dst, coef);

    // 3) readout of layer-0 features
    hg_readout_kernel<<<(N_NODES * 16 + 255) / 256, 256, 0, stream>>>(x, gids, r0);

    // 4) layer 1: h1 = scatter_sum(x[src] * coef, dst); readout -> r1
    hg_scatter_kernel<<<(N_EDGES * 16) / 256, 256, 0, stream>>>(x, coef, src, dst, h1);
    hg_readout_kernel<<<(N_NODES * 16 + 255) / 256, 256, 0, stream>>>(h1, gids, r1);

    // 5) layer 2: h2 = scatter_sum(h1[src] * coef, dst); readout -> r2
    hg_scatter_kernel<<<(N_EDGES * 16) / 256, 256, 0, stream>>>(h1, coef, src, dst, h2);
    hg_readout_kernel<<<(N_NODES * 16 + 255) / 256, 256, 0, stream>>>(h2, gids, r2);

    // 6) epilogue GEMM + bias + leaky_relu via WMMA f32 16x16x4
    hg_final_wmma_kernel<<<64, 32, 0, stream>>>(r0, r1, r2, W, bvec, out);
}